// Detection1D_87376814670262
// MI455X (gfx1250) — compile-verified
//
#include <hip/hip_runtime.h>
#include <cstdint>

// ---------------- problem constants (fixed by the reference) ----------------
#define NQ      131072          // proposals per batch
#define TOPK    10
#define NEGV    (-1e30f)
#define OKTH    (-5e29f)        // "ok" threshold = NEG/2
#define BLOCKT  1024            // 32 waves of 32
#define TILE4   (BLOCKT * 4)    // 4096 elements staged per step
#define NSTEP   (NQ / TILE4)    // 32 steps -> 128 elems per thread-chunk

typedef float v4f __attribute__((ext_vector_type(4)));
typedef float v2f __attribute__((ext_vector_type(2)));

// ---- CDNA5 async global->LDS path via inline asm (gfx1250 only) ------------
#if defined(__HIP_DEVICE_COMPILE__) && defined(__gfx1250__)
#define HAS_ASYNC 1
#endif

#ifdef HAS_ASYNC
// GLOBAL_LOAD_ASYNC_TO_LDS_B128: VDST = VGPR with LDS byte address,
// VADDR = VGPR pair with 64-bit global address, SADDR = off.
// Low 32 bits of a generic LDS pointer are the LDS byte offset (flat aperture
// truncation rule, ISA 10.2).
__device__ __forceinline__ void async_copy16(const void* g, void* l) {
  unsigned int       lo = (unsigned int)(uintptr_t)l;
  unsigned long long ga = (unsigned long long)(uintptr_t)g;
  asm volatile("global_load_async_to_lds_b128 %0, %1, off"
               :
               : "v"(lo), "v"(ga)
               : "memory");
}
__device__ __forceinline__ void wait_async_le1() {
  asm volatile("s_wait_asynccnt 1" ::: "memory");
}
__device__ __forceinline__ void wait_async_0() {
  asm volatile("s_wait_asynccnt 0" ::: "memory");
}
#endif

// ---------------- helpers ----------------
__device__ __forceinline__ unsigned int f32_sortable(float f) {
  unsigned int u = __float_as_uint(f);
  return (u & 0x80000000u) ? ~u : (u | 0x80000000u);
}
__device__ __forceinline__ float sortable_f32(unsigned int u) {
  return __uint_as_float((u & 0x80000000u) ? (u ^ 0x80000000u) : ~u);
}
// max(key) == argmax score, ties -> lowest index (matches jnp.argmax)
__device__ __forceinline__ unsigned long long mkkey(float s, int idx) {
  return ((unsigned long long)f32_sortable(s) << 32) |
         (unsigned long long)(unsigned int)(0x7FFFFFFF - idx);
}

// ---------------- kernel 1: decode + validity mask into L2-resident SoA -----
__global__ __launch_bounds__(256) void decode_kernel(
    const v2f* __restrict__ clf, const v4f* __restrict__ reg,
    const v4f* __restrict__ bxs, v2f* __restrict__ ox1,
    v2f* __restrict__ ox2, v2f* __restrict__ osc) {
  int t = blockIdx.x * 256 + threadIdx.x;       // two elements per thread
  v2f sc = __builtin_nontemporal_load(&clf[t]); // inputs are read-once: NT
  v4f rg = __builtin_nontemporal_load(&reg[t]);
  v4f bb = __builtin_nontemporal_load(&bxs[t]);
  v2f X1, X2, S;
#pragma unroll
  for (int e = 0; e < 2; ++e) {
    float b1 = bb[2 * e], b2 = bb[2 * e + 1];
    float w  = b2 - b1;
    float pc = b1 + 0.5f * w + (rg[2 * e] * 0.1f) * w;
    float pw = expf(rg[2 * e + 1] * 0.2f) * w;
    float x1 = fminf(fmaxf(pc - 0.5f * pw, 0.0f), 416.0f);
    float x2 = fminf(fmaxf(pc + 0.5f * pw, 0.0f), 416.0f);
    X1[e] = x1; X2[e] = x2;
    S[e] = ((sc[e] > 0.01f) && ((x2 - x1) > 3.0f)) ? sc[e] : NEGV;
  }
  ox1[t] = X1; ox2[t] = X2; osc[t] = S;
}

// element fetch: scratch SoA (fast path) or inline re-decode (fallback)
template <bool USE_WS>
__device__ __forceinline__ void fetch_elem(
    const float* __restrict__ sG, const float* __restrict__ x1G,
    const float* __restrict__ x2G, const float* __restrict__ clf,
    const float* __restrict__ reg, const float* __restrict__ bxs,
    size_t base, int i, float& s, float& X1, float& X2) {
  size_t g = base + (size_t)i;
  if constexpr (USE_WS) {
    s = sG[g]; X1 = x1G[g]; X2 = x2G[g];
  } else {
    float b1 = bxs[2 * g], b2 = bxs[2 * g + 1];
    float w  = b2 - b1;
    float pc = b1 + 0.5f * w + (reg[2 * g] * 0.1f) * w;
    float pw = expf(reg[2 * g + 1] * 0.2f) * w;
    X1 = fminf(fmaxf(pc - 0.5f * pw, 0.0f), 416.0f);
    X2 = fminf(fmaxf(pc + 0.5f * pw, 0.0f), 416.0f);
    float sc = clf[g];
    s = ((sc > 0.01f) && ((X2 - X1) > 3.0f)) ? sc : NEGV;
  }
}

// ---------------- kernel 2: incremental top-10 NMS, one block per batch -----
template <bool USE_WS>
__global__ __launch_bounds__(BLOCKT) void nms_kernel(
    const float* __restrict__ sG, const float* __restrict__ x1G,
    const float* __restrict__ x2G, const float* __restrict__ clf,
    const float* __restrict__ reg, const float* __restrict__ bxs,
    float* __restrict__ out) {
  const int b   = blockIdx.x;
  const int tid = threadIdx.x;
  const size_t base = (size_t)b * NQ;

  __shared__ float selX1[TOPK];
  __shared__ float selX2[TOPK];
  __shared__ int   nselS;
  __shared__ int   flagS;
  __shared__ unsigned long long warpRed[BLOCKT / 32];
  __shared__ unsigned long long winnerS;
  extern __shared__ float sbuf[];     // [2][TILE4] double buffer (USE_WS path)

  if (tid == 0) nselS = 0;
  __syncthreads();

  // per-thread chunk state lives in registers across all 10 NMS steps
  unsigned long long ckey = 0ULL;     // chunk winner key (0 = empty)
  float wx1 = -1.0f, wx2 = -1.0f;     // chunk winner box

  // ---- pass 0: chunk-argmax over the score array ----
  if constexpr (USE_WS) {
    const float* sB = sG + base;
#ifdef HAS_ASYNC
    // CDNA5 async global->LDS staging, double-buffered, per-wave ASYNCcnt sync
    async_copy16(sB + (tid << 2), sbuf + (tid << 2));
    for (int j = 0; j < NSTEP; ++j) {
      if (j + 1 < NSTEP) {
        async_copy16(sB + (size_t)(j + 1) * TILE4 + (tid << 2),
                     sbuf + ((j + 1) & 1) * TILE4 + (tid << 2));
        wait_async_le1();              // tile j has landed in LDS
      } else {
        wait_async_0();
      }
      v4f sv = reinterpret_cast<const v4f*>(sbuf + (j & 1) * TILE4)[tid];
      int gi = j * TILE4 + (tid << 2);
#pragma unroll
      for (int l = 0; l < 4; ++l) {
        unsigned long long kk = mkkey(sv[l], gi + l);
        if (kk > ckey) ckey = kk;
      }
    }
#else
    for (int j = 0; j < NSTEP; ++j) {
      v4f sv = reinterpret_cast<const v4f*>(sB)[j * BLOCKT + tid];
      int gi = j * TILE4 + (tid << 2);
#pragma unroll
      for (int l = 0; l < 4; ++l) {
        unsigned long long kk = mkkey(sv[l], gi + l);
        if (kk > ckey) ckey = kk;
      }
    }
#endif
    int widx = 0x7FFFFFFF - (int)(unsigned int)(ckey & 0xFFFFFFFFull);
    wx1 = x1G[base + widx];
    wx2 = x2G[base + widx];
  } else {
    for (int j = 0; j < NSTEP; ++j) {
      int gi = j * TILE4 + (tid << 2);
#pragma unroll
      for (int l = 0; l < 4; ++l) {
        float s, X1, X2;
        fetch_elem<false>(sG, x1G, x2G, clf, reg, bxs, base, gi + l, s, X1, X2);
        unsigned long long kk = mkkey(s, gi + l);
        if (kk > ckey) { ckey = kk; wx1 = X1; wx2 = X2; }
      }
    }
  }

  // ---- 10 sequential selections ----
  for (int k = 0; k < TOPK; ++k) {
    // block-wide argmax of chunk keys: wave32 shfl tree, then 32-wave LDS tree
    unsigned long long key = ckey;
#pragma unroll
    for (int off = 16; off > 0; off >>= 1) {
      unsigned long long o = __shfl_xor(key, off, 32);
      if (o > key) key = o;
    }
    if ((tid & 31) == 0) warpRed[tid >> 5] = key;
    __syncthreads();
    if (tid < 32) {
      unsigned long long kk = warpRed[tid];
#pragma unroll
      for (int off = 16; off > 0; off >>= 1) {
        unsigned long long o = __shfl_xor(kk, off, 32);
        if (o > kk) kk = o;
      }
      if (tid == 0) { winnerS = kk; flagS = 0; }
    }
    __syncthreads();

    unsigned long long wkey = winnerS;
    float wsc = sortable_f32((unsigned int)(wkey >> 32));
    bool ok = (wkey != 0ULL) && (wsc > OKTH);

    if (ok && ckey == wkey) {      // unique owner thread (keys embed index)
      int ns = nselS;
      selX1[ns] = wx1; selX2[ns] = wx2;
      nselS = ns + 1;
      flagS = 1;
      float* o = out + ((size_t)b * TOPK + k) * 3;
      o[0] = wx1; o[1] = wx2; o[2] = wsc;
    }
    if (!ok && tid == 0) {
      float* o = out + ((size_t)b * TOPK + k) * 3;
      o[0] = -1.0f; o[1] = -1.0f; o[2] = -1.0f;
    }
    __syncthreads();

    // rescan only chunks whose register-resident winner got suppressed
    if (flagS != 0 && (k + 1) < TOPK) {
      int ns = nselS;
      float nX1 = selX1[ns - 1], nX2 = selX2[ns - 1];
      bool need_rescan = false;
      if (ckey != 0ULL) {
        float it = fmaxf(fminf(wx2, nX2) - fmaxf(wx1, nX1), 0.0f);
        float un = (wx2 - wx1) + (nX2 - nX1) - it + 1e-9f;
        need_rescan = it > 0.5f * un;   // iou > NMS_T (covers self-suppression)
      }
      if (need_rescan) {
        ckey = 0ULL;
        for (int j = 0; j < NSTEP; ++j) {
          int gi = j * TILE4 + (tid << 2);
#pragma unroll
          for (int l = 0; l < 4; ++l) {
            int i = gi + l;
            float s, X1, X2;
            fetch_elem<USE_WS>(sG, x1G, x2G, clf, reg, bxs, base, i, s, X1, X2);
            if (s <= OKTH) continue;
            bool sup = false;
            for (int m = 0; m < ns; ++m) {
              float it2 = fmaxf(fminf(X2, selX2[m]) - fmaxf(X1, selX1[m]), 0.0f);
              float un2 = (X2 - X1) + (selX2[m] - selX1[m]) - it2 + 1e-9f;
              if (it2 > 0.5f * un2) { sup = true; break; }
            }
            if (!sup) {
              unsigned long long kk = mkkey(s, i);
              if (kk > ckey) { ckey = kk; wx1 = X1; wx2 = X2; }
            }
          }
        }
      }
    }
    __syncthreads();
  }
}

// ---------------- host launcher ----------------
extern "C" void kernel_launch(void* const* d_in, const int* in_sizes, int n_in,
                              void* d_out, int out_size, void* d_ws, size_t ws_size,
                              hipStream_t stream) {
  const float* clf = (const float*)d_in[0];
  const float* reg = (const float*)d_in[1];
  const float* bxs = (const float*)d_in[2];
  float* out = (float*)d_out;

  const int B = in_sizes[0] / NQ;                 // 64
  const size_t total = (size_t)B * NQ;            // 8,388,608
  const size_t need  = total * sizeof(float) * 3; // ~96 MB (L2-resident)

  if (ws_size >= need) {
    float* x1 = (float*)d_ws;
    float* x2 = x1 + total;
    float* s  = x2 + total;
    const int thr = 256;
    const int blocks = (int)(total / 2 / thr);    // 2 elems per thread
    decode_kernel<<<blocks, thr, 0, stream>>>(
        (const v2f*)clf, (const v4f*)reg, (const v4f*)bxs,
        (v2f*)x1, (v2f*)x2, (v2f*)s);
    nms_kernel<true><<<B, BLOCKT, 2 * TILE4 * (int)sizeof(float), stream>>>(
        s, x1, x2, clf, reg, bxs, out);
  } else {
    nms_kernel<false><<<B, BLOCKT, 0, stream>>>(
        nullptr, nullptr, nullptr, clf, reg, bxs, out);
  }
}